// CLAHETransform_23914377904404
// MI455X (gfx1250) — compile-verified
//
#include <hip/hip_runtime.h>

typedef float v2f __attribute__((ext_vector_type(2)));
typedef float v4f __attribute__((ext_vector_type(4)));
typedef float v8f __attribute__((ext_vector_type(8)));

#define IMG_H 2048
#define IMG_W 2048
#define HW (IMG_H * IMG_W)
#define GH 8
#define GW 8
#define TH 256
#define TW 256
#define NB 256
#define PIXELS (TH * TW)      // 65536
#define MAXV 10240.0f         // int(40.0 * 65536 // 256)
#define ROWS_PER_BLOCK 16
#define K1_THREADS 1024
#define K1_WAVES (K1_THREADS / 32)

__device__ __forceinline__ int pix_bin(float v) {
  int b = (int)((v / 255.0f) * 256.0f);  // matches (img/255 * 256).astype(int32)
  return min(max(b, 0), NB - 1);
}

// ---------------------------------------------------------------------------
// Kernel 1: one workgroup per tile (192 blocks x 1024 threads = 32 waves).
//   b128 tile reads -> 32 wave-private LDS histograms (ds_add_u32) ->
//   clip -> redistribute -> 256-bin inclusive cumsum via
//   V_WMMA_F32_16X16X4_F32 (Hm x U + Ls x T triangular matmuls) -> LUT.
// ---------------------------------------------------------------------------
__global__ __launch_bounds__(K1_THREADS) void clahe_hist_lut(
    const float* __restrict__ img, float* __restrict__ luts) {
  __shared__ unsigned int s_hist[K1_WAVES][NB];  // 32 KB wave-private histograms
  __shared__ float s_hc[NB];                     // clipped+redistributed histogram
  __shared__ float s_sum;                        // sum of clipped histogram

  const int tid = threadIdx.x;
  const int t = blockIdx.x;                      // tile id: c*64 + ty*8 + tx
  const int c  = t >> 6;
  const int ty = (t >> 3) & 7;
  const int tx = t & 7;
  const int wid = tid >> 5;                      // wave id (wave32)

  for (int i = tid; i < K1_WAVES * NB; i += K1_THREADS)
    ((unsigned int*)s_hist)[i] = 0u;
  if (tid == 0) s_sum = 0.0f;
  __syncthreads();

  // --- histogram: 16 rows per iteration, float4 per thread, coalesced ---
  const v4f* base4 = (const v4f*)(img + (size_t)c * HW
                                  + (size_t)(ty * TH) * IMG_W + (size_t)(tx * TW));
  const int W4 = IMG_W / 4;
  const int rowofs = tid >> 6;                   // 0..15
  const int c4 = tid & 63;                       // float4 column within tile row
  for (int it = 0; it < TH / 16; ++it) {
    int rr = it * 16 + rowofs;
    v4f v = base4[(size_t)rr * W4 + c4];
    #pragma unroll
    for (int j = 0; j < 4; ++j)
      atomicAdd(&s_hist[wid][pix_bin(v[j])], 1u);   // ds_add_u32
  }
  __syncthreads();

  // --- reduce copies, clip, sum (integer-valued floats: order-independent) ---
  if (tid < NB) {
    unsigned int h = 0;
    #pragma unroll
    for (int w = 0; w < K1_WAVES; ++w) h += s_hist[w][tid];
    float hmin = fminf((float)h, MAXV);
    s_hc[tid] = hmin;
    atomicAdd(&s_sum, hmin);                     // ds_add_f32
  }
  __syncthreads();

  if (tid < NB) {
    float clipped  = (float)PIXELS - s_sum;
    float residual = fmodf(clipped, 256.0f);
    float redist   = (clipped - residual) * (1.0f / 256.0f);
    s_hc[tid] = s_hc[tid] + redist + (((float)tid < residual) ? 1.0f : 0.0f);
  }
  __syncthreads();

  // --- 256-element inclusive scan as triangular matmuls on WMMA ---
  // Hm[m][n] = hc[16m+n]; cum[16m+n] = (Hm x U)[m][n] + (Ls x T)[m][n]
  // Every wave computes redundantly (EXEC all-1s), identical stores.
  const int lane = tid & 31;
  const int col  = lane & 15;
  const int half = lane >> 4;                    // 0: lanes 0-15, 1: lanes 16-31
  const int m    = lane & 15;                    // A-matrix row for this lane

  v8f acc = {};
  // Step 1: P = Hm x U  (U[k][n] = 1 iff k <= n), K=16 as 4 chunks of K=4
  #pragma unroll
  for (int q = 0; q < 4; ++q) {
    int k0 = 4 * q + 2 * half;
    int k1 = k0 + 1;
    v2f a; a.x = s_hc[m * 16 + k0]; a.y = s_hc[m * 16 + k1];
    v2f b; b.x = (k0 <= col) ? 1.0f : 0.0f; b.y = (k1 <= col) ? 1.0f : 0.0f;
    acc = __builtin_amdgcn_wmma_f32_16x16x4_f32(false, a, false, b,
                                                (short)0, acc, false, false);
  }

  // Row totals tot[r] = P[r][15] live in lanes 15 (r<8) and 31 (r>=8)
  float tot[16];
  #pragma unroll
  for (int i = 0; i < 8; ++i) {
    tot[i]     = __shfl(acc[i], 15, 32);
    tot[i + 8] = __shfl(acc[i], 31, 32);
  }

  // Step 2: acc += Ls x T  (Ls[m][r] = 1 iff r < m; T[r][n] = tot[r])
  #pragma unroll
  for (int q = 0; q < 4; ++q) {
    int k0 = 4 * q + 2 * half;
    int k1 = k0 + 1;
    v2f a; a.x = (k0 < m) ? 1.0f : 0.0f; a.y = (k1 < m) ? 1.0f : 0.0f;
    v2f b;
    b.x = half ? tot[4 * q + 2] : tot[4 * q + 0];
    b.y = half ? tot[4 * q + 3] : tot[4 * q + 1];
    acc = __builtin_amdgcn_wmma_f32_16x16x4_f32(false, a, false, b,
                                                (short)0, acc, false, false);
  }

  // --- LUT = floor(clip(cumsum * 255/65536, 0, 255)) ---
  const float lut_scale = 255.0f / 65536.0f;     // exact in fp32
  #pragma unroll
  for (int i = 0; i < 8; ++i) {
    int row = i + 8 * half;                      // C/D layout: VGPR i -> M=i / M=i+8
    int idx = row * 16 + col;
    float lv = floorf(fminf(fmaxf(acc[i] * lut_scale, 0.0f), 255.0f));
    luts[(size_t)t * NB + idx] = lv;             // duplicate identical stores OK
  }
}

// ---------------------------------------------------------------------------
// Kernel 2: apply. One block = 16 rows of one channel; full channel LUT
// (64 tiles x 256 bins = 64 KB) cached in LDS; b128 image/mask/output path,
// 4 LDS gathers per pixel.
// ---------------------------------------------------------------------------
__global__ __launch_bounds__(256) void clahe_apply(const float* __restrict__ img,
                                                   const float* __restrict__ mask,
                                                   const float* __restrict__ luts,
                                                   float* __restrict__ out) {
  __shared__ float s_lut[GH * GW * NB];          // 16384 floats = 64 KB
  const int tid = threadIdx.x;
  const int c = blockIdx.y;
  const int y_base = blockIdx.x * ROWS_PER_BLOCK;
  const int W4 = IMG_W / 4;

  const float* lsrc = luts + (size_t)c * GH * GW * NB;
  for (int i = tid; i < GH * GW * NB; i += 256) s_lut[i] = lsrc[i];
  __syncthreads();

  for (int r = 0; r < ROWS_PER_BLOCK; ++r) {
    int y = y_base + r;
    float ysf = ((float)y + 0.5f) / (float)TH - 0.5f;
    float yf = floorf(ysf);
    float wy = ysf - yf;
    float omwy = 1.0f - wy;
    int iy0 = min(max((int)yf, 0), GH - 1);
    int iy1 = min(max((int)yf + 1, 0), GH - 1);
    int rowbase0 = iy0 * GW;
    int rowbase1 = iy1 * GW;

    const v4f* irow4 = (const v4f*)(img  + (size_t)c * HW + (size_t)y * IMG_W);
    const v4f* mrow4 = (const v4f*)(mask + (size_t)y * IMG_W);
    v4f*       orow4 = (v4f*)      (out  + (size_t)c * HW + (size_t)y * IMG_W);

    for (int x4 = tid; x4 < W4; x4 += 256) {     // 2 iterations per row
      v4f v = irow4[x4];
      v4f mv = mrow4[x4];
      v4f o;
      #pragma unroll
      for (int j = 0; j < 4; ++j) {
        int x = x4 * 4 + j;
        int b = pix_bin(v[j]);

        float xsf = ((float)x + 0.5f) / (float)TW - 0.5f;
        float xf = floorf(xsf);
        float wx = xsf - xf;
        float omwx = 1.0f - wx;
        int ix0 = min(max((int)xf, 0), GW - 1);
        int ix1 = min(max((int)xf + 1, 0), GW - 1);

        float l00 = s_lut[(rowbase0 + ix0) * NB + b];
        float l01 = s_lut[(rowbase0 + ix1) * NB + b];
        float l10 = s_lut[(rowbase1 + ix0) * NB + b];
        float l11 = s_lut[(rowbase1 + ix1) * NB + b];

        float val = l00 * omwy * omwx + l01 * omwy * wx
                  + l10 * wy   * omwx + l11 * wy   * wx;
        o[j] = floorf(val) * mv[j];
      }
      orow4[x4] = o;
    }
  }
}

// ---------------------------------------------------------------------------
// Kernel 3: tail of the output tuple: label, keypoints, mask, probe.
// ---------------------------------------------------------------------------
__global__ __launch_bounds__(256) void clahe_tail(const int* __restrict__ label,
                                                  const float* __restrict__ kp,
                                                  const float* __restrict__ mask,
                                                  const int* __restrict__ probe,
                                                  float* __restrict__ out) {
  size_t i = (size_t)blockIdx.x * blockDim.x + threadIdx.x;
  const size_t O = (size_t)3 * HW;
  const size_t n = 1 + 8 + (size_t)HW + 1;
  if (i >= n) return;
  if (i == 0)                  out[O] = (float)label[0];
  else if (i < 9)              out[O + i] = kp[i - 1];
  else if (i < 9 + (size_t)HW) out[O + i] = mask[i - 9];
  else                         out[O + i] = (float)probe[0];
}

extern "C" void kernel_launch(void* const* d_in, const int* in_sizes, int n_in,
                              void* d_out, int out_size, void* d_ws, size_t ws_size,
                              hipStream_t stream) {
  const float* img   = (const float*)d_in[0];
  const int*   label = (const int*)d_in[1];
  const float* kp    = (const float*)d_in[2];
  const float* mask  = (const float*)d_in[3];
  const int*   probe = (const int*)d_in[4];
  float* out  = (float*)d_out;
  float* luts = (float*)d_ws;                    // 192 * 256 floats = 192 KB scratch

  clahe_hist_lut<<<dim3(3 * GH * GW), dim3(K1_THREADS), 0, stream>>>(img, luts);
  clahe_apply<<<dim3(IMG_H / ROWS_PER_BLOCK, 3), dim3(256), 0, stream>>>(img, mask, luts, out);

  size_t n_tail = 1 + 8 + (size_t)HW + 1;
  clahe_tail<<<dim3((unsigned)((n_tail + 255) / 256)), dim3(256), 0, stream>>>(
      label, kp, mask, probe, out);
}